// MatchingMatrixComp_81827716923963
// MI455X (gfx1250) — compile-verified
//
#include <hip/hip_runtime.h>
#include <cstdint>

#define LOG2E 1.44269504088896340736f

typedef unsigned int u32x4 __attribute__((ext_vector_type(4)));
typedef int          i32x8 __attribute__((ext_vector_type(8)));
typedef int          i32x4 __attribute__((ext_vector_type(4)));

constexpr int NM = 128;       // matrix is NM x NM
constexpr int STRIDE = 129;   // padded LDS row stride (conflict-free rows AND cols)

__global__ __launch_bounds__(256) void sinkhorn_kernel(
    const float* __restrict__ X, const int* __restrict__ itersp,
    float* __restrict__ out)
{
    __shared__ float sm[NM * STRIDE];   // 66,048 B padded tile
    __shared__ float red_m[256];
    __shared__ float red_s[256];

    const int tid  = threadIdx.x;
    const int L    = tid & 127;   // line index (row in row-pass, col in col-pass)
    const int h    = tid >> 7;    // which half of the line this thread owns
    const int base = h * 64;
    const int p    = blockIdx.x;

    const float* src = X   + (size_t)p * (NM * NM);
    float*       dst = out + (size_t)p * (NM * NM);

    // ---- Tensor Data Mover: DMA 128x128 f32 tile global->LDS, padding each
    // ---- 128-DWORD row with 1 DWORD so the LDS stride becomes 129 floats.
    if (tid < 32) {
        unsigned long long ga = (unsigned long long)(uintptr_t)src;
        unsigned lds_base = (unsigned)(uintptr_t)&sm[0];

        u32x4 g0;
        g0[0] = 1u;                                   // count=1 (valid), user mode
        g0[1] = lds_base;                             // lds_addr
        g0[2] = (unsigned)(ga & 0xFFFFFFFFu);         // global_addr[31:0]
        g0[3] = (unsigned)((ga >> 32) & 0x01FFFFFFu)  // global_addr[56:32]
              | (2u << 30);                           // type = 2 ("image")

        i32x8 g1;
        g1[0] = (2 << 16)                 // data_size = 4 bytes
              | (1 << 20)                 // pad_enable
              | (6 << 22)                 // pad_interval: 2^(6+1)=128 DWORDs
              | (0 << 25);                // pad_amount: 1 DWORD
        g1[1] = (NM & 0xFFFF) << 16;      // tensor_dim0 lo16 = 128
        g1[2] = (NM & 0xFFFF) << 16;      // dim0 hi16=0 | tensor_dim1 lo16 = 128
        g1[3] = (NM & 0xFFFF) << 16;      // dim1 hi16=0 | tile_dim0 = 128
        g1[4] = NM;                       // tile_dim1 = 128 | tile_dim2 = 0
        g1[5] = NM;                       // tensor_dim0_stride = 128
        g1[6] = (NM * NM) << 16;          // d0str hi16=0 | tensor_dim1_stride lo16
        g1[7] = 0;

        i32x4 g2;
        g2[0] = 1;          // tensor_dim2 = 1 (benign for 2D tile)
        g2[1] = 1;          // tensor_dim3 = 1
        g2[2] = 0;
        g2[3] = 0;          // tile_dim3 = 0 (unused)
        i32x4 g3;
        g3[0] = 0;
        g3[1] = 1 << 16;    // tensor_dim4 = 1
        g3[2] = 0;
        g3[3] = 0;          // tile_dim4 = 0 (unused)

        i32x8 g4;           // 5th descriptor group (clang-23 6-arg form); zero-filled
        g4[0] = 0; g4[1] = 0; g4[2] = 0; g4[3] = 0;
        g4[4] = 0; g4[5] = 0; g4[6] = 0; g4[7] = 0;

        __builtin_amdgcn_tensor_load_to_lds(g0, g1, g2, g3, g4, 0);
    }
    __builtin_amdgcn_s_wait_tensorcnt(0);
    __syncthreads();

    const int iters = itersp[0];

    for (int it = 0; it < iters; ++it) {
        // ================= row softmax (axis = -1) =================
        {
            float e[64];
            float mb = -3.402823466e38f;     // max over raw values
            #pragma unroll
            for (int j = 0; j < 64; ++j) {
                float x = sm[L * STRIDE + base + j];
                e[j] = x;
                mb = fmaxf(mb, x);
            }
            float nb2 = -mb * LOG2E;
            float s = 0.0f;
            #pragma unroll
            for (int j = 0; j < 64; ++j) {
                // exp(x - mb) == exp2(fma(x, log2e, -mb*log2e)): 1 fma + 1 trans
                e[j] = __builtin_amdgcn_exp2f(__builtin_fmaf(e[j], LOG2E, nb2));
                s += e[j];
            }
            red_m[tid] = mb; red_s[tid] = s;
            __syncthreads();                 // also separates LDS reads from writes
            float mo = red_m[tid ^ 128], so = red_s[tid ^ 128];
            float MB = fmaxf(mb, mo);
            float fs = __builtin_amdgcn_exp2f((mb - MB) * LOG2E);
            float fo = __builtin_amdgcn_exp2f((mo - MB) * LOG2E);
            float scale = fs / (s * fs + so * fo);
            #pragma unroll
            for (int j = 0; j < 64; ++j)
                sm[L * STRIDE + base + j] = e[j] * scale;
            __syncthreads();
        }
        // ================= column softmax (axis = -2) ==============
        {
            const bool last = (it == iters - 1);
            float e[64];
            float mb = -3.402823466e38f;
            #pragma unroll
            for (int j = 0; j < 64; ++j) {
                float x = sm[(base + j) * STRIDE + L];
                e[j] = x;
                mb = fmaxf(mb, x);
            }
            float nb2 = -mb * LOG2E;
            float s = 0.0f;
            #pragma unroll
            for (int j = 0; j < 64; ++j) {
                e[j] = __builtin_amdgcn_exp2f(__builtin_fmaf(e[j], LOG2E, nb2));
                s += e[j];
            }
            red_m[tid] = mb; red_s[tid] = s;
            __syncthreads();
            float mo = red_m[tid ^ 128], so = red_s[tid ^ 128];
            float MB = fmaxf(mb, mo);
            float fs = __builtin_amdgcn_exp2f((mb - MB) * LOG2E);
            float fo = __builtin_amdgcn_exp2f((mo - MB) * LOG2E);
            float scale = fs / (s * fs + so * fo);
            if (last) {
                // column ownership => lanes hit consecutive addresses: coalesced
                #pragma unroll
                for (int j = 0; j < 64; ++j)
                    dst[(base + j) * NM + L] = e[j] * scale;
            } else {
                #pragma unroll
                for (int j = 0; j < 64; ++j)
                    sm[(base + j) * STRIDE + L] = e[j] * scale;
                __syncthreads();
            }
        }
    }
}

extern "C" void kernel_launch(void* const* d_in, const int* in_sizes, int n_in,
                              void* d_out, int out_size, void* d_ws, size_t ws_size,
                              hipStream_t stream) {
    const float* X     = (const float*)d_in[0];
    const int*   iters = (const int*)d_in[1];
    float*       out   = (float*)d_out;
    int P = in_sizes[0] / (NM * NM);
    sinkhorn_kernel<<<dim3(P), dim3(256), 0, stream>>>(X, iters, out);
}